// NCF_plus_44616120270976
// MI455X (gfx1250) — compile-verified
//
#include <hip/hip_runtime.h>
#include <hip/hip_bf16.h>

typedef __attribute__((ext_vector_type(2))) float v2f;
typedef __attribute__((ext_vector_type(8))) float v8f;

#define TILE_B   64
#define NTHREADS 128   // 4 wave32s
#define STR_H    196   // 192 + 4 pad (16B-aligned rows, 4-bank skew)
#define STR_H1   68    // 64 + 4 pad
#define STR_W1   196
#define STR_W2   68

// LDS floats: sH(64*196) sH1(64*68) sW1(64*196) sW2(96*68) sB1(64) sB2(96) sWo(192) sPart(64)
#define LDS_FLOATS (TILE_B*STR_H + TILE_B*STR_H1 + 64*STR_W1 + 96*STR_W2 + 64 + 96 + 192 + TILE_B)

__global__ void ncf_fused_kernel(
    const int*  __restrict__ user_ids,
    const int*  __restrict__ movie_ids,
    const int*  __restrict__ genre_ids,
    const float* __restrict__ user_mem_w,
    const float* __restrict__ movie_mem_w,
    const float* __restrict__ ue_gmf,
    const float* __restrict__ me_gmf,
    const float* __restrict__ ge_gmf,
    const float* __restrict__ ue_mlp,
    const float* __restrict__ me_mlp,
    const float* __restrict__ ge_mlp,
    const float* __restrict__ w1,
    const float* __restrict__ b1,
    const float* __restrict__ w2,
    const float* __restrict__ b2,
    const float* __restrict__ w_out,
    const float* __restrict__ b_out,
    float* __restrict__ out)
{
    extern __shared__ float smem[];
    float* sH    = smem;                       // [64][196]
    float* sH1   = sH    + TILE_B * STR_H;     // [64][68]
    float* sW1   = sH1   + TILE_B * STR_H1;    // [64][196]
    float* sW2   = sW1   + 64 * STR_W1;        // [96][68]
    float* sB1   = sW2   + 96 * STR_W2;        // [64]
    float* sB2   = sB1   + 64;                 // [96]
    float* sWo   = sB2   + 96;                 // [192]
    float* sPart = sWo   + 192;                // [64]

    const int tid = threadIdx.x;
    const int b0  = blockIdx.x * TILE_B;

    // ---------------- Phase 1: stage weights + gather h tile + GMF partial ----------------

    // w1 [64][192] -> sW1 [64][196]
    for (int i = tid; i < 64 * 48; i += NTHREADS) {
        int n = i / 48, k4 = i % 48;
        *(float4*)&sW1[n * STR_W1 + k4 * 4] = ((const float4*)w1)[i];
    }
    // w2 [96][64] -> sW2 [96][68]
    for (int i = tid; i < 96 * 16; i += NTHREADS) {
        int n = i / 16, k4 = i % 16;
        *(float4*)&sW2[n * STR_W2 + k4 * 4] = ((const float4*)w2)[i];
    }
    for (int i = tid; i < 64;  i += NTHREADS) sB1[i] = b1[i];
    for (int i = tid; i < 96;  i += NTHREADS) sB2[i] = b2[i];
    for (int i = tid; i < 192; i += NTHREADS) sWo[i] = w_out[i];

    // h[:, 0:96]  = ue_mlp[user]
    for (int i = tid; i < TILE_B * 24; i += NTHREADS) {
        int r = i / 24, c4 = i % 24;
        size_t u = (size_t)user_ids[b0 + r];
        *(float4*)&sH[r * STR_H + c4 * 4] = *(const float4*)&ue_mlp[u * 96 + c4 * 4];
    }
    // h[:, 96:160] = me_mlp[movie]
    for (int i = tid; i < TILE_B * 16; i += NTHREADS) {
        int r = i / 16, c4 = i % 16;
        size_t m = (size_t)movie_ids[b0 + r];
        *(float4*)&sH[r * STR_H + 96 + c4 * 4] = *(const float4*)&me_mlp[m * 64 + c4 * 4];
    }
    // h[:, 160:192] = mean of 3 genre rows (32-wide)
    for (int i = tid; i < TILE_B * 8; i += NTHREADS) {
        int r = i / 8, c4 = i % 8;
        const int* g = &genre_ids[(b0 + r) * 3];
        float4 a = *(const float4*)&ge_mlp[g[0] * 32 + c4 * 4];
        float4 c = *(const float4*)&ge_mlp[g[1] * 32 + c4 * 4];
        float4 d = *(const float4*)&ge_mlp[g[2] * 32 + c4 * 4];
        float4 s;
        s.x = (a.x + c.x + d.x) * (1.0f / 3.0f);
        s.y = (a.y + c.y + d.y) * (1.0f / 3.0f);
        s.z = (a.z + c.z + d.z) * (1.0f / 3.0f);
        s.w = (a.w + c.w + d.w) * (1.0f / 3.0f);
        *(float4*)&sH[r * STR_H + 160 + c4 * 4] = s;
    }

    // GMF branch + memory biases, reduced against w_out[0:96] immediately
    if (tid < TILE_B) {
        int b = b0 + tid;
        size_t u = (size_t)user_ids[b];
        size_t m = (size_t)movie_ids[b];
        const int* g = &genre_ids[b * 3];
        float acc = user_mem_w[u] + movie_mem_w[m] + b_out[0];
        const float* ug = &ue_gmf[u * 96];
        const float* mg = &me_gmf[m * 64];
        const float* g0 = &ge_gmf[g[0] * 32];
        const float* g1 = &ge_gmf[g[1] * 32];
        const float* g2 = &ge_gmf[g[2] * 32];
        #pragma unroll 4
        for (int j = 0; j < 64; ++j) acc += ug[j] * mg[j] * w_out[j];
        #pragma unroll 4
        for (int j = 0; j < 32; ++j) {
            float gm = (g0[j] + g1[j] + g2[j]) * (1.0f / 3.0f);
            acc += ug[64 + j] * gm * w_out[64 + j];
        }
        sPart[tid] = acc;
    }

    __syncthreads();

    // ---------------- Phase 2: MLP via V_WMMA_F32_16X16X4_F32 ----------------
    const int lane  = tid & 31;
    const int wv    = tid >> 5;          // wave id: 16-row tile
    const int rbase = wv * 16;
    const int lrow  = lane & 15;         // row (A) / col-in-tile (B,C)
    const int khalf = (lane >> 4) * 2;   // K sub-pair per half-wave
    const int vrow0 = (lane >> 4) * 8;   // C/D row base per half-wave

    const v8f zero8 = {0.f, 0.f, 0.f, 0.f, 0.f, 0.f, 0.f, 0.f};

    // GEMM1: D[16x64] = h[16x192] x w1^T[192x64]
    v8f acc1[4] = {zero8, zero8, zero8, zero8};
    {
        const float* aRow = &sH[(rbase + lrow) * STR_H];
        for (int k0 = 0; k0 < 192; k0 += 4) {
            v2f a = *(const v2f*)&aRow[k0 + khalf];
            #pragma unroll
            for (int nt = 0; nt < 4; ++nt) {
                int n = nt * 16 + lrow;
                v2f bb = *(const v2f*)&sW1[n * STR_W1 + k0 + khalf];  // B[k][n] = w1[n][k]
                acc1[nt] = __builtin_amdgcn_wmma_f32_16x16x4_f32(
                    false, a, false, bb, (short)0, acc1[nt], false, false);
            }
        }
    }
    // bias + ReLU -> sH1
    #pragma unroll
    for (int nt = 0; nt < 4; ++nt) {
        int n = nt * 16 + lrow;
        float bias = sB1[n];
        #pragma unroll
        for (int v = 0; v < 8; ++v) {
            float x = acc1[nt][v] + bias;
            sH1[(rbase + vrow0 + v) * STR_H1 + n] = x > 0.f ? x : 0.f;
        }
    }
    __syncthreads();

    // GEMM2: D[16x96] = h1[16x64] x w2^T[64x96]
    v8f acc2[6] = {zero8, zero8, zero8, zero8, zero8, zero8};
    {
        const float* aRow = &sH1[(rbase + lrow) * STR_H1];
        for (int k0 = 0; k0 < 64; k0 += 4) {
            v2f a = *(const v2f*)&aRow[k0 + khalf];
            #pragma unroll
            for (int nt = 0; nt < 6; ++nt) {
                int n = nt * 16 + lrow;
                v2f bb = *(const v2f*)&sW2[n * STR_W2 + k0 + khalf];  // B[k][n] = w2[n][k]
                acc2[nt] = __builtin_amdgcn_wmma_f32_16x16x4_f32(
                    false, a, false, bb, (short)0, acc2[nt], false, false);
            }
        }
    }

    // bias + ReLU fused with dot against w_out[96:192], then 16-lane reduction
    float part[8] = {0.f, 0.f, 0.f, 0.f, 0.f, 0.f, 0.f, 0.f};
    #pragma unroll
    for (int nt = 0; nt < 6; ++nt) {
        int n = nt * 16 + lrow;
        float bias = sB2[n];
        float wo   = sWo[96 + n];
        #pragma unroll
        for (int v = 0; v < 8; ++v) {
            float x = acc2[nt][v] + bias;
            x = x > 0.f ? x : 0.f;
            part[v] += x * wo;
        }
    }
    #pragma unroll
    for (int msk = 1; msk < 16; msk <<= 1) {
        #pragma unroll
        for (int v = 0; v < 8; ++v) part[v] += __shfl_xor(part[v], msk, 32);
    }
    if (lrow == 0) {
        #pragma unroll
        for (int v = 0; v < 8; ++v) {
            int r = rbase + vrow0 + v;
            out[b0 + r] = sPart[r] + part[v];
        }
    }
}

extern "C" void kernel_launch(void* const* d_in, const int* in_sizes, int n_in,
                              void* d_out, int out_size, void* d_ws, size_t ws_size,
                              hipStream_t stream) {
    const int*   user_ids    = (const int*)d_in[0];
    const int*   movie_ids   = (const int*)d_in[1];
    const int*   genre_ids   = (const int*)d_in[2];
    // d_in[3] = offsets (arange*3) — implied by GENRES_PER_MOVIE=3, unused
    const float* user_mem_w  = (const float*)d_in[4];
    const float* movie_mem_w = (const float*)d_in[5];
    const float* ue_gmf      = (const float*)d_in[6];
    const float* me_gmf      = (const float*)d_in[7];
    const float* ge_gmf      = (const float*)d_in[8];
    const float* ue_mlp      = (const float*)d_in[9];
    const float* me_mlp      = (const float*)d_in[10];
    const float* ge_mlp      = (const float*)d_in[11];
    const float* w1          = (const float*)d_in[12];
    const float* b1          = (const float*)d_in[13];
    const float* w2          = (const float*)d_in[14];
    const float* b2          = (const float*)d_in[15];
    const float* w_out       = (const float*)d_in[16];
    const float* b_out       = (const float*)d_in[17];
    float* out = (float*)d_out;

    const int B = in_sizes[0];
    const int grid = B / TILE_B;
    const size_t ldsBytes = (size_t)LDS_FLOATS * sizeof(float);

    ncf_fused_kernel<<<grid, NTHREADS, ldsBytes, stream>>>(
        user_ids, movie_ids, genre_ids,
        user_mem_w, movie_mem_w,
        ue_gmf, me_gmf, ge_gmf,
        ue_mlp, me_mlp, ge_mlp,
        w1, b1, w2, b2, w_out, b_out, out);
}